// unfold_block_gcn_50113678409886
// MI455X (gfx1250) — compile-verified
//
#include <hip/hip_runtime.h>
#include <math.h>

typedef __attribute__((ext_vector_type(2))) float v2f;
typedef __attribute__((ext_vector_type(8))) float v8f;

// ---------------------------------------------------------------------------
// Degree / normalization
// ---------------------------------------------------------------------------
__global__ __launch_bounds__(256) void k_deg_init(float* __restrict__ deg, int n) {
  int i = blockIdx.x * blockDim.x + threadIdx.x;
  if (i < n) deg[i] = 1.0f;  // self-loop weight
}

__global__ __launch_bounds__(256) void k_deg_edges(const int* __restrict__ dst,
                                                   const float* __restrict__ ew,
                                                   float* __restrict__ deg, int e) {
  int i = blockIdx.x * blockDim.x + threadIdx.x;
  if (i < e) atomicAdd(&deg[dst[i]], ew[i]);
}

__global__ __launch_bounds__(256) void k_rsqrt_inplace(float* __restrict__ d, int n) {
  int i = blockIdx.x * blockDim.x + threadIdx.x;
  if (i < n) d[i] = rsqrtf(d[i]);  // deg >= 1 always (self loops)
}

// ---------------------------------------------------------------------------
// Dense transform via V_WMMA_F32_16X16X4_F32, compile-time KIN so the K-loop
// fully unrolls into a branch-free WMMA chain. W ([KIN x 64]) staged in LDS,
// pair-swizzled so a lane's B-fragment {W[k0][col], W[k0+1][col]} is one
// aligned 64-bit LDS load. Rows >= KIN are zero-padded.
//
// Epilogue: the 16x64 result tile is round-tripped through LDS (group-XOR
// swizzle to dodge bank aliasing) so each lane owns ONE output row and 32
// contiguous columns -> single bounds predicate, float4 global stores.
//
// Computes xw = act(in) @ W  (in stride == KIN) and initializes the
// aggregation accumulator acc = xw * dinv^2 + bias (self-loop + bias folded).
// One wave owns a 16-row x 64-col tile (4 accumulator chains).
// ---------------------------------------------------------------------------
template <int KIN, bool RELU>
__global__ __launch_bounds__(256)
void k_gemm_wmma(const float* __restrict__ in,
                 const float* __restrict__ W,     // [KIN x 64] row-major
                 const float* __restrict__ bias,  // [64]
                 const float* __restrict__ dinv,  // [n]
                 float* __restrict__ xw,          // [n x 64]
                 float* __restrict__ acc,         // [n x 64]
                 int n)
{
  constexpr int KSTEPS = (KIN + 3) / 4;
  constexpr int KP     = KSTEPS * 4;              // padded K
  __shared__ float Wl[KP * 64];                   // pair-swizzled weights
  __shared__ float Bl[64];                        // bias
  __shared__ float Tile[8 * 16 * 64];             // per-wave 16x64 result

  // Stage + swizzle W into LDS (zero-pad rows >= KIN), stage bias.
  for (int idx = threadIdx.x; idx < KP * 64; idx += 256) {
    const int k = idx >> 6, col = idx & 63;
    float v = 0.f;
    if (k < KIN) v = W[k * 64 + col];
    Wl[(k >> 1) * 128 + col * 2 + (k & 1)] = v;
  }
  if (threadIdx.x < 64) Bl[threadIdx.x] = bias[threadIdx.x];
  __syncthreads();

  const int lane = threadIdx.x & 31;
  const int wave = threadIdx.x >> 5;
  const int row0 = (blockIdx.x * 8 + wave) * 16;  // 16 rows per wave
  if (row0 >= n) return;                          // uniform per wave

  const int half = lane >> 4;                     // 0: lanes 0-15, 1: 16-31
  const int lid  = lane & 15;
  int m = row0 + lid;                             // A-fragment source row
  if (m >= n) m = n - 1;                          // clamp (stores are guarded)
  const float* __restrict__ arow = in + (size_t)m * KIN;

  v8f c[4] = {};

#pragma unroll
  for (int ks = 0; ks < KSTEPS; ++ks) {
    const int k0 = ks * 4 + half * 2;             // this lane supplies K=k0,k0+1
    v2f a;
    if constexpr ((KIN & 3) == 0) {
      a = *(const v2f*)(arow + k0);               // aligned 8B (k0 even)
    } else {
      const int ka0 = (k0     < KIN) ? k0     : 0;
      const int ka1 = (k0 + 1 < KIN) ? k0 + 1 : 0;
      float a0 = arow[ka0];
      float a1 = arow[ka1];
      a.x = (k0     < KIN) ? a0 : 0.f;            // branch-free selects
      a.y = (k0 + 1 < KIN) ? a1 : 0.f;
    }
    if (RELU) { a.x = fmaxf(a.x, 0.f); a.y = fmaxf(a.y, 0.f); }

#pragma unroll
    for (int t = 0; t < 4; ++t) {
      const int col = t * 16 + lid;
      const v2f b = *(const v2f*)(&Wl[(k0 >> 1) * 128 + col * 2]);
      c[t] = __builtin_amdgcn_wmma_f32_16x16x4_f32(
          /*neg_a=*/false, a, /*neg_b=*/false, b,
          /*c_mod=*/(short)0, c[t], /*reuse_a=*/false, /*reuse_b=*/false);
    }
  }

  // --- Transpose tile through LDS (group-XOR swizzle on 4-float groups) ---
  float* tp = &Tile[wave * 1024];
#pragma unroll
  for (int t = 0; t < 4; ++t) {
#pragma unroll
    for (int r = 0; r < 8; ++r) {
      const int row_l = r + half * 8;             // local row 0..15
      const int col   = t * 16 + lid;
      const int grp   = ((col >> 2) ^ row_l) & 15;
      tp[row_l * 64 + grp * 4 + (col & 3)] = c[t][r];
    }
  }

  // --- Readback: lane owns row (lane>>1), 8 float4 groups of columns ------
  const int   row_l = lane >> 1;                  // 0..15
  const int   row   = row0 + row_l;
  if (row < n) {                                  // ONE predicate per lane
    const float di = dinv[row];
    const float dd = di * di;
    float* __restrict__ xwrow  = xw  + (size_t)row * 64;
    float* __restrict__ accrow = acc + (size_t)row * 64;
#pragma unroll
    for (int j = 0; j < 8; ++j) {
      const int g  = (lane & 1) * 8 + j;          // logical group (cols g*4..+3)
      const int gs = (g ^ row_l) & 15;            // swizzled group
      const float4 v  = *(const float4*)(tp + row_l * 64 + gs * 4);
      const float4 b4 = *(const float4*)(&Bl[g * 4]);
      float4 a4;
      a4.x = v.x * dd + b4.x;
      a4.y = v.y * dd + b4.y;
      a4.z = v.z * dd + b4.z;
      a4.w = v.w * dd + b4.w;
      *(float4*)(xwrow  + g * 4) = v;             // xw
      *(float4*)(accrow + g * 4) = a4;            // acc = xw*dinv^2 + bias
    }
  }
}

// ---------------------------------------------------------------------------
// Edge aggregation, 64-dim messages: 16 lanes per edge, float4 per lane.
// acc[dst] += xw[src] * (dinv[src]*ew*dinv[dst])
// ---------------------------------------------------------------------------
__global__ __launch_bounds__(256)
void k_edge_agg64(const int* __restrict__ src, const int* __restrict__ dst,
                  const float* __restrict__ ew, const float* __restrict__ dinv,
                  const float* __restrict__ xw, float* __restrict__ acc, int e)
{
  const int tid  = blockIdx.x * blockDim.x + threadIdx.x;
  const int idx  = tid >> 4;
  if (idx >= e) return;
  const int part = tid & 15;
  const int s = src[idx], d = dst[idx];
  const float norm = dinv[s] * ew[idx] * dinv[d];
  const float4 v = *(const float4*)(xw + (size_t)s * 64 + part * 4);
  float* out = acc + (size_t)d * 64 + part * 4;
  atomicAdd(out + 0, v.x * norm);
  atomicAdd(out + 1, v.y * norm);
  atomicAdd(out + 2, v.z * norm);
  atomicAdd(out + 3, v.w * norm);
}

// 1-dim messages (last conv layer)
__global__ __launch_bounds__(256)
void k_edge_agg1(const int* __restrict__ src, const int* __restrict__ dst,
                 const float* __restrict__ ew, const float* __restrict__ dinv,
                 const float* __restrict__ xw1, float* __restrict__ acc1, int e)
{
  const int i = blockIdx.x * blockDim.x + threadIdx.x;
  if (i >= e) return;
  const int s = src[i], d = dst[i];
  atomicAdd(&acc1[d], xw1[s] * dinv[s] * ew[i] * dinv[d]);
}

// ---------------------------------------------------------------------------
// Conv 64 -> 1: xw1 = relu(hin) . W2 ; acc1 = xw1*dinv^2 + b2
// ---------------------------------------------------------------------------
__global__ __launch_bounds__(256)
void k_dot64(const float* __restrict__ hin, const float* __restrict__ W2,
             const float* __restrict__ b2, const float* __restrict__ dinv,
             float* __restrict__ xw1, float* __restrict__ acc1, int n)
{
  const int i = blockIdx.x * blockDim.x + threadIdx.x;
  if (i >= n) return;
  const float4* row = (const float4*)(hin + (size_t)i * 64);
  float s = 0.f;
#pragma unroll
  for (int k = 0; k < 16; ++k) {
    float4 v = row[k];
    v.x = fmaxf(v.x, 0.f); v.y = fmaxf(v.y, 0.f);
    v.z = fmaxf(v.z, 0.f); v.w = fmaxf(v.w, 0.f);
    s += v.x * W2[k*4+0] + v.y * W2[k*4+1] + v.z * W2[k*4+2] + v.w * W2[k*4+3];
  }
  xw1[i] = s;
  const float di = dinv[i];
  acc1[i] = s * di * di + b2[0];
}

// Tower-O tail: x_sol = h2*Wl + bl ; build xin = concat(x, x_sol)
__global__ __launch_bounds__(256)
void k_lin_concat(const float* __restrict__ acc1, const float* __restrict__ Wl,
                  const float* __restrict__ bl, const float* __restrict__ x,
                  float* __restrict__ xsol, float* __restrict__ xin, int n)
{
  const int i = blockIdx.x * blockDim.x + threadIdx.x;
  if (i >= n) return;
  const float v = acc1[i] * Wl[0] + bl[0];
  xsol[i] = v;
  const float4 xi = *(const float4*)(x + (size_t)i * 4);
  xin[(size_t)i*5+0] = xi.x;
  xin[(size_t)i*5+1] = xi.y;
  xin[(size_t)i*5+2] = xi.z;
  xin[(size_t)i*5+3] = xi.w;
  xin[(size_t)i*5+4] = v;
}

// Tower-G tail: gamma = sigmoid(h2*Wl+bl); x_new = x_last + gamma*(x_sol-x_last)
__global__ __launch_bounds__(256)
void k_final(const float* __restrict__ acc1, const float* __restrict__ Wl,
             const float* __restrict__ bl, const float* __restrict__ x,
             const float* __restrict__ xsol, float* __restrict__ out, int n)
{
  const int i = blockIdx.x * blockDim.x + threadIdx.x;
  if (i >= n) return;
  const float g = acc1[i] * Wl[0] + bl[0];
  const float gamma = 1.f / (1.f + expf(-g));
  const float xl = x[(size_t)i*4+3];
  out[i]     = xl + gamma * (xsol[i] - xl);
  out[n + i] = gamma;
}

// ---------------------------------------------------------------------------
extern "C" void kernel_launch(void* const* d_in, const int* in_sizes, int n_in,
                              void* d_out, int out_size, void* d_ws, size_t ws_size,
                              hipStream_t stream)
{
  const float* x    = (const float*)d_in[0];
  const int*   ei   = (const int*)  d_in[1];
  const float* ew   = (const float*)d_in[2];
  const float* o_W0 = (const float*)d_in[3];
  const float* o_b0 = (const float*)d_in[4];
  const float* o_W1 = (const float*)d_in[5];
  const float* o_b1 = (const float*)d_in[6];
  const float* o_W2 = (const float*)d_in[7];
  const float* o_b2 = (const float*)d_in[8];
  const float* o_Wl = (const float*)d_in[9];
  const float* o_bl = (const float*)d_in[10];
  const float* g_W0 = (const float*)d_in[11];
  const float* g_b0 = (const float*)d_in[12];
  const float* g_W1 = (const float*)d_in[13];
  const float* g_b1 = (const float*)d_in[14];
  const float* g_W2 = (const float*)d_in[15];
  const float* g_b2 = (const float*)d_in[16];
  const float* g_Wl = (const float*)d_in[17];
  const float* g_bl = (const float*)d_in[18];

  const int N = in_sizes[0] / 4;     // x is [N,4]
  const int E = in_sizes[2];         // edge_weights is [E]
  const int* src = ei;
  const int* dst = ei + E;

  float* ws   = (float*)d_ws;
  float* dinv = ws;                           // [N] (deg computed in place)
  float* xw   = dinv + N;                     // [N*64]
  float* accA = xw   + (size_t)N * 64;        // [N*64]
  float* accB = accA + (size_t)N * 64;        // [N*64]
  float* xin  = accB + (size_t)N * 64;        // [N*5]
  float* xsol = xin  + (size_t)N * 5;         // [N]
  float* acc1 = xsol + N;                     // [N]
  float* xw1  = acc1 + N;                     // [N]

  float* out = (float*)d_out;

  const int TB = 256;
  const int gN   = (N + TB - 1) / TB;
  const int gE   = (E + TB - 1) / TB;
  const int gE16 = (int)(((long long)E * 16 + TB - 1) / TB);
  const int gG   = (N + 16 * 8 - 1) / (16 * 8);   // gemm: 8 waves x 16 rows

  // Degree normalization (shared by every conv layer)
  k_deg_init <<<gN, TB, 0, stream>>>(dinv, N);
  k_deg_edges<<<gE, TB, 0, stream>>>(dst, ew, dinv, E);
  k_rsqrt_inplace<<<gN, TB, 0, stream>>>(dinv, N);

  // ---------------- Tower O (optim): 4 -> 64 -> 64 -> 1 -------------------
  k_gemm_wmma<4, false><<<gG, TB, 0, stream>>>(x,    o_W0, o_b0, dinv, xw, accA, N);
  k_edge_agg64<<<gE16, TB, 0, stream>>>(src, dst, ew, dinv, xw, accA, E);
  k_gemm_wmma<64, true><<<gG, TB, 0, stream>>>(accA, o_W1, o_b1, dinv, xw, accB, N);
  k_edge_agg64<<<gE16, TB, 0, stream>>>(src, dst, ew, dinv, xw, accB, E);
  k_dot64<<<gN, TB, 0, stream>>>(accB, o_W2, o_b2, dinv, xw1, acc1, N);
  k_edge_agg1<<<gE, TB, 0, stream>>>(src, dst, ew, dinv, xw1, acc1, E);
  k_lin_concat<<<gN, TB, 0, stream>>>(acc1, o_Wl, o_bl, x, xsol, xin, N);

  // ---------------- Tower G (gamma): 5 -> 64 -> 64 -> 1 -------------------
  k_gemm_wmma<5, false><<<gG, TB, 0, stream>>>(xin,  g_W0, g_b0, dinv, xw, accA, N);
  k_edge_agg64<<<gE16, TB, 0, stream>>>(src, dst, ew, dinv, xw, accA, E);
  k_gemm_wmma<64, true><<<gG, TB, 0, stream>>>(accA, g_W1, g_b1, dinv, xw, accB, N);
  k_edge_agg64<<<gE16, TB, 0, stream>>>(src, dst, ew, dinv, xw, accB, E);
  k_dot64<<<gN, TB, 0, stream>>>(accB, g_W2, g_b2, dinv, xw1, acc1, N);
  k_edge_agg1<<<gE, TB, 0, stream>>>(src, dst, ew, dinv, xw1, acc1, E);
  k_final<<<gN, TB, 0, stream>>>(acc1, g_Wl, g_bl, x, xsol, out, N);
}